// GroupRigidParamVelocityStrategy_87239375716835
// MI455X (gfx1250) — compile-verified
//
#include <hip/hip_runtime.h>
#include <hip/hip_bf16.h>
#include <math.h>

// ---------------- problem constants ----------------
constexpr int CB = 32;     // batch
constexpr int CN = 512;    // keypoints
constexpr int CD = 1024;   // token dim
constexpr int CG = 16;     // groups
constexpr int CS = 32;     // group size
constexpr int CE = 1024;   // edges
constexpr int CH = 256;    // hidden
constexpr float CEPS = 1e-6f;
constexpr float TAU_MIN = 1e-3f;

// ---------------- workspace layout (float offsets) ----------------
constexpr size_t OFF_GTOK   = 0;                       // 512*1024
constexpr size_t OFF_H      = 524288;                  // 512*256
constexpr size_t OFF_PARAMS = 655360;                  // 512*9
constexpr size_t OFF_R      = 659968;                  // 512*9
constexpr size_t OFF_T      = 664576;                  // 512*3
constexpr size_t OFF_XSUM   = 666112;                  // 32*512*3
constexpr size_t OFF_COUNTS = 715264;                  // 512
constexpr size_t OFF_V      = 715776;                  // 32*512*3
constexpr size_t OFF_DX4    = 764928;                  // 32*1024*4 (padded dx, w=0)
constexpr size_t OFF_RHS    = 896000;                  // 32*1024
constexpr size_t OFF_LAM    = 928768;                  // 32*1024
constexpr size_t OFF_SIGN   = 961536;                  // 1024*1024
constexpr size_t OFF_CORR   = 2010112;                 // 32*512*3
constexpr size_t OFF_A      = 2059264;                 // 32*1024*1024

typedef float v2f __attribute__((ext_vector_type(2)));
typedef float v8f __attribute__((ext_vector_type(8)));
typedef unsigned int u32x4 __attribute__((ext_vector_type(4)));
typedef int i32x4 __attribute__((ext_vector_type(4)));
typedef int i32x8 __attribute__((ext_vector_type(8)));

__device__ inline v8f wmma_f32_k4(v2f a, v2f b, v8f c) {
  // D(16x16,f32) = A(16x4,f32) x B(4x16,f32) + C  — CDNA5 V_WMMA_F32_16X16X4_F32
  return __builtin_amdgcn_wmma_f32_16x16x4_f32(false, a, false, b, (short)0, c,
                                               false, false);
}

#if defined(__has_builtin)
#if __has_builtin(__builtin_amdgcn_tensor_load_to_lds) && \
    __has_builtin(__builtin_amdgcn_s_wait_tensorcnt)
#define GRP_HAVE_TDM 1
#endif
#endif

#ifdef GRP_HAVE_TDM
// Issue a TDM descriptor: 16x16 f32 tile at `gaddr` (row stride 1024 floats,
// tensor 1024x1024) -> LDS offset 0, padded +1 DWORD per 16-DWORD row so it
// lands in a float[16][17] LDS array. Wave-uniform; call from one wave.
__device__ inline void grp_tdm_load_diag(const float* gaddr) {
  unsigned long long ga = (unsigned long long)(size_t)gaddr;
  u32x4 g0 = {1u,                                            // count=1, user mode
              0u,                                            // lds_addr = 0
              (unsigned)(ga & 0xffffffffull),                // global addr lo
              (unsigned)((ga >> 32) & 0x01ffffffu) | (2u << 30)};  // addr hi | type=2
  i32x8 g1 = {(int)0x00D20000,   // data_size=4B, pad_enable, pad_interval=16DW, pad=1DW
              (int)0x04000000,   // tensor_dim0 = 1024 (bits 79:48)
              (int)0x04000000,   // tensor_dim1 = 1024 (bits 111:80)
              (int)0x00100000,   // tile_dim0 = 16 (bits 127:112)
              (int)0x00000010,   // tile_dim1 = 16, tile_dim2 = 0
              (int)0x00000400,   // tensor_dim0_stride = 1024
              0,                 // dim0_stride hi, dim1_stride lo
              (int)0x00000010};  // tensor_dim1_stride = 1M (bits 255:224)
  i32x4 g2 = {};
  i32x4 g3 = {};
#if defined(__clang_major__) && (__clang_major__ >= 23)
  i32x8 g4 = {};
  __builtin_amdgcn_tensor_load_to_lds(g0, g1, g2, g3, g4, 0);
#else
  __builtin_amdgcn_tensor_load_to_lds(g0, g1, g2, g3, 0);
#endif
  __builtin_amdgcn_s_wait_tensorcnt(0);
}
#endif

// ---------------- 0: zero a float range ----------------
__global__ void grp_zero(float* p, int n) {
  int i = blockIdx.x * blockDim.x + threadIdx.x;
  if (i < n) p[i] = 0.0f;
}

// ---------------- 1: group token mean ----------------
__global__ void grp_group_mean(const float* __restrict__ tok,
                               const int* __restrict__ groups,
                               float* __restrict__ gt) {
  int bg = blockIdx.x;          // b*G+g
  int b = bg / CG, g = bg % CG;
  for (int d = threadIdx.x; d < CD; d += blockDim.x) {
    float sum = 0.0f;
    for (int s = 0; s < CS; ++s) {
      int t = groups[g * CS + s];
      sum += tok[((size_t)b * CN + t) * CD + d];
    }
    gt[(size_t)bg * CD + d] = sum * (1.0f / (float)CS);
  }
}

// ---------------- 2: h = silu(g_tok @ W1 + b1) via fp32 WMMA ----------------
__global__ __launch_bounds__(256) void grp_gemm1_wmma(
    const float* __restrict__ gt, const float* __restrict__ W1,
    const float* __restrict__ b1, float* __restrict__ h) {
  int wid = blockIdx.x * 8 + (threadIdx.x >> 5);  // 512 waves = 32x16 tiles
  int lane = threadIdx.x & 31;
  int lhalf = lane >> 4, l16 = lane & 15;
  int m0 = (wid >> 4) * 16;
  int n0 = (wid & 15) * 16;
  const float* arow = gt + (size_t)(m0 + l16) * CD;
  const float* bcol = W1 + (n0 + l16);
  v8f acc = {};
  for (int k = 0; k < CD; k += 4) {
    __builtin_prefetch(&arow[k + 64], 0, 0);   // global_prefetch_b8
    v2f a, bb;
    a.x = arow[k + 2 * lhalf];
    a.y = arow[k + 2 * lhalf + 1];
    bb.x = bcol[(size_t)(k + 2 * lhalf) * CH];
    bb.y = bcol[(size_t)(k + 2 * lhalf + 1) * CH];
    acc = wmma_f32_k4(a, bb, acc);
  }
  float bias = b1[n0 + l16];
  for (int r = 0; r < 8; ++r) {
    int row = m0 + r + 8 * lhalf;
    float x = acc[r] + bias;
    float sig = 1.0f / (1.0f + expf(-x));
    h[(size_t)row * CH + n0 + l16] = x * sig;
  }
}

// ---------------- 3: params = h @ W2 + b2 (tiny) ----------------
__global__ void grp_gemm2(const float* __restrict__ h, const float* __restrict__ W2,
                          const float* __restrict__ b2, float* __restrict__ params) {
  int idx = blockIdx.x * blockDim.x + threadIdx.x;
  if (idx >= CB * CG * 9) return;
  int m = idx / 9, o = idx % 9;
  float acc = b2[o];
  for (int k = 0; k < CH; ++k) acc += h[(size_t)m * CH + k] * W2[k * 9 + o];
  params[idx] = acc;
}

// ---------------- 4: r6d -> rotation, translation ----------------
__global__ void grp_rot(const float* __restrict__ params, float* __restrict__ R,
                        float* __restrict__ T) {
  int idx = blockIdx.x * blockDim.x + threadIdx.x;
  if (idx >= CB * CG) return;
  const float* p = params + idx * 9;
  float a1x = p[0], a1y = p[1], a1z = p[2];
  float a2x = p[3], a2y = p[4], a2z = p[5];
  float n1 = fmaxf(sqrtf(a1x * a1x + a1y * a1y + a1z * a1z), 1e-9f);
  float b1x = a1x / n1, b1y = a1y / n1, b1z = a1z / n1;
  float d = b1x * a2x + b1y * a2y + b1z * a2z;
  float ux = a2x - d * b1x, uy = a2y - d * b1y, uz = a2z - d * b1z;
  float n2 = fmaxf(sqrtf(ux * ux + uy * uy + uz * uz), 1e-9f);
  float b2x = ux / n2, b2y = uy / n2, b2z = uz / n2;
  float b3x = b1y * b2z - b1z * b2y;
  float b3y = b1z * b2x - b1x * b2z;
  float b3z = b1x * b2y - b1y * b2x;
  float* Rm = R + idx * 9;
  Rm[0] = b1x; Rm[1] = b2x; Rm[2] = b3x;
  Rm[3] = b1y; Rm[4] = b2y; Rm[5] = b3y;
  Rm[6] = b1z; Rm[7] = b2z; Rm[8] = b3z;
  T[idx * 3 + 0] = p[6]; T[idx * 3 + 1] = p[7]; T[idx * 3 + 2] = p[8];
}

// ---------------- 5: rigid transform + scatter-add ----------------
__global__ void grp_scatter(const float* __restrict__ R, const float* __restrict__ T,
                            const float* __restrict__ tmpl, const int* __restrict__ groups,
                            float* __restrict__ xsum, float* __restrict__ counts) {
  int idx = blockIdx.x * blockDim.x + threadIdx.x;
  if (idx >= CB * CG * CS) return;
  int b = idx / (CG * CS);
  int r = idx % (CG * CS);
  int g = r / CS, s = r % CS;
  int t = groups[g * CS + s];
  const float* Rm = R + (b * CG + g) * 9;
  const float* tv = T + (b * CG + g) * 3;
  float Y0 = tmpl[t * 3 + 0], Y1 = tmpl[t * 3 + 1], Y2 = tmpl[t * 3 + 2];
  for (int i = 0; i < 3; ++i) {
    float Xi = Rm[i * 3 + 0] * Y0 + Rm[i * 3 + 1] * Y1 + Rm[i * 3 + 2] * Y2 + tv[i];
    atomicAdd(&xsum[((size_t)b * CN + t) * 3 + i], Xi);
  }
  if (b == 0) atomicAdd(&counts[t], 1.0f);
}

// ---------------- 6: v = (x_goal - kp)/tau ----------------
__global__ void grp_vel(const float* __restrict__ xsum, const float* __restrict__ counts,
                        const float* __restrict__ kp, const float* __restrict__ ts,
                        float* __restrict__ v) {
  int idx = blockIdx.x * blockDim.x + threadIdx.x;
  if (idx >= CB * CN) return;
  int b = idx / CN, n = idx % CN;
  float cnt = fmaxf(counts[n], 1.0f);
  float tau = fmaxf(1.0f - ts[b], TAU_MIN);
  for (int c = 0; c < 3; ++c) {
    size_t o = (size_t)idx * 3 + c;
    v[o] = (xsum[o] / cnt - kp[o]) / tau;
  }
}

// ---------------- 7: per-edge dx (padded), rhs ----------------
__global__ void grp_edges(const float* __restrict__ kp, const float* __restrict__ v,
                          const int* __restrict__ ei, float* __restrict__ dx4,
                          float* __restrict__ rhs) {
  int idx = blockIdx.x * blockDim.x + threadIdx.x;
  if (idx >= CB * CE) return;
  int b = idx / CE, e = idx % CE;
  int i = ei[e], j = ei[CE + e];
  float r = 0.0f;
  for (int c = 0; c < 3; ++c) {
    float dxc = kp[((size_t)b * CN + i) * 3 + c] - kp[((size_t)b * CN + j) * 3 + c];
    float dvc = v[((size_t)b * CN + i) * 3 + c] - v[((size_t)b * CN + j) * 3 + c];
    dx4[(size_t)idx * 4 + c] = dxc;
    r += dxc * dvc;
  }
  dx4[(size_t)idx * 4 + 3] = 0.0f;
  rhs[idx] = 2.0f * r;
}

// ---------------- 8: sign matrix (batch-independent) ----------------
__global__ void grp_sign(const int* __restrict__ ei, float* __restrict__ sgn) {
  int idx = blockIdx.x * blockDim.x + threadIdx.x;
  if (idx >= CE * CE) return;
  int e = idx / CE, f = idx % CE;
  int ie = ei[e], je = ei[CE + e];
  int if_ = ei[f], jf = ei[CE + f];
  float s = (float)(ie == if_) - (float)(ie == jf) - (float)(je == if_) + (float)(je == jf);
  sgn[idx] = s;
}

// ---------------- 9: A = 4*(dx dx^T)*sign + eps*I via fp32 WMMA ----------------
__global__ __launch_bounds__(256) void grp_buildA(const float* __restrict__ dx4,
                                                  const float* __restrict__ sgn,
                                                  float* __restrict__ A) {
  int wid = blockIdx.x * 8 + (threadIdx.x >> 5);  // 32 batches * 64*64 tiles
  int lane = threadIdx.x & 31;
  int lhalf = lane >> 4, l16 = lane & 15;
  int b = wid >> 12;
  int rem = wid & 4095;
  int e0 = (rem >> 6) * 16;
  int f0 = (rem & 63) * 16;
  const float* dxb = dx4 + (size_t)b * CE * 4;
  v2f a, bb;
  a.x = dxb[(size_t)(e0 + l16) * 4 + 2 * lhalf];
  a.y = dxb[(size_t)(e0 + l16) * 4 + 2 * lhalf + 1];
  bb.x = dxb[(size_t)(f0 + l16) * 4 + 2 * lhalf];
  bb.y = dxb[(size_t)(f0 + l16) * 4 + 2 * lhalf + 1];
  v8f acc = {};
  acc = wmma_f32_k4(a, bb, acc);
  float* Ab = A + (size_t)b * CE * CE;
  for (int r = 0; r < 8; ++r) {
    int e = e0 + r + 8 * lhalf, f = f0 + l16;
    float val = 4.0f * acc[r] * sgn[(size_t)e * CE + f];
    if (e == f) val += CEPS;
    Ab[(size_t)e * CE + f] = val;
  }
}

// ---------------- 10: blocked right-looking batched Cholesky ----------------
// One 8-wave workgroup per batch. Diagonal 16x16 tile fetched by the Tensor
// Data Mover into LDS (padded row stride 17 via TDM pad feature); diag factor
// + TRSM scalar; trailing SYRK C -= P_i P_j^T with fp32 WMMA (A-frag negated).
__global__ __launch_bounds__(256) void grp_chol(float* __restrict__ A) {
  int b = blockIdx.x;
  float* Ab = A + (size_t)b * CE * CE;
  __shared__ float sd[16][17];
  int tid = threadIdx.x;
  int wave = tid >> 5, lane = tid & 31;
  int lhalf = lane >> 4, l16 = lane & 15;
  for (int kb = 0; kb < 64; ++kb) {
    int kc = kb * 16;
#ifdef GRP_HAVE_TDM
    if (wave == 0) {  // one wave issues the tensor DMA, waits TENSORcnt
      grp_tdm_load_diag(Ab + (size_t)kc * CE + kc);
    }
    __syncthreads();
#else
    {  // fallback: manual load of diagonal block
      int r = tid >> 4, c = tid & 15;
      sd[r][c] = Ab[(size_t)(kc + r) * CE + kc + c];
    }
    __syncthreads();
#endif
    for (int k = 0; k < 16; ++k) {  // unblocked Cholesky of 16x16
      if (tid == 0) sd[k][k] = sqrtf(sd[k][k]);
      __syncthreads();
      if (tid > k && tid < 16) sd[tid][k] /= sd[k][k];
      __syncthreads();
      {
        int r = tid >> 4, c = tid & 15;
        if (r > k && c > k && c <= r) sd[r][c] -= sd[r][k] * sd[c][k];
      }
      __syncthreads();
    }
    {  // write back L diag, zero strict upper
      int r = tid >> 4, c = tid & 15;
      Ab[(size_t)(kc + r) * CE + kc + c] = (c <= r) ? sd[r][c] : 0.0f;
    }
    __syncthreads();
    // panel TRSM: rows below solve against L_diag^T
    for (int i = kc + 16 + tid; i < CE; i += 256) {
      float a[16];
      for (int c = 0; c < 16; ++c) a[c] = Ab[(size_t)i * CE + kc + c];
      for (int c = 0; c < 16; ++c) {
        float t = a[c];
        for (int m = 0; m < c; ++m) t -= a[m] * sd[c][m];
        a[c] = t / sd[c][c];
      }
      for (int c = 0; c < 16; ++c) Ab[(size_t)i * CE + kc + c] = a[c];
    }
    __threadfence();
    __syncthreads();
    // trailing update (WMMA): tiles (ib,jb), kb<jb<=ib, wave-uniform assignment
    int t = 0;
    for (int ib = kb + 1; ib < 64; ++ib) {
      for (int jb = kb + 1; jb <= ib; ++jb, ++t) {
        if ((t & 7) != wave) continue;
        int gi = ib * 16, gj = jb * 16;
        v8f acc;
        for (int r = 0; r < 8; ++r)
          acc[r] = Ab[(size_t)(gi + r + 8 * lhalf) * CE + gj + l16];
        for (int kk = 0; kk < 16; kk += 4) {
          v2f a, bb;
          a.x = -Ab[(size_t)(gi + l16) * CE + kc + kk + 2 * lhalf];
          a.y = -Ab[(size_t)(gi + l16) * CE + kc + kk + 2 * lhalf + 1];
          bb.x = Ab[(size_t)(gj + l16) * CE + kc + kk + 2 * lhalf];
          bb.y = Ab[(size_t)(gj + l16) * CE + kc + kk + 2 * lhalf + 1];
          acc = wmma_f32_k4(a, bb, acc);  // C - P_i P_j^T
        }
        for (int r = 0; r < 8; ++r)
          Ab[(size_t)(gi + r + 8 * lhalf) * CE + gj + l16] = acc[r];
      }
    }
    __threadfence();
    __syncthreads();
  }
}

// ---------------- 11: forward/backward substitution per batch ----------------
__global__ __launch_bounds__(256) void grp_solve(const float* __restrict__ A,
                                                 const float* __restrict__ rhs,
                                                 float* __restrict__ lam) {
  int b = blockIdx.x;
  const float* L = A + (size_t)b * CE * CE;
  __shared__ float ys[CE];
  __shared__ float ls[CE];
  __shared__ float red[256];
  int tid = threadIdx.x;
  for (int e = 0; e < CE; ++e) {  // L y = rhs
    float s = 0.0f;
    for (int f = tid; f < e; f += 256) s += L[(size_t)e * CE + f] * ys[f];
    red[tid] = s;
    __syncthreads();
    for (int st = 128; st > 0; st >>= 1) {
      if (tid < st) red[tid] += red[tid + st];
      __syncthreads();
    }
    if (tid == 0) ys[e] = (rhs[(size_t)b * CE + e] - red[0]) / L[(size_t)e * CE + e];
    __syncthreads();
  }
  for (int e = CE - 1; e >= 0; --e) {  // L^T lam = y
    float s = 0.0f;
    for (int f = e + 1 + tid; f < CE; f += 256) s += L[(size_t)f * CE + e] * ls[f];
    red[tid] = s;
    __syncthreads();
    for (int st = 128; st > 0; st >>= 1) {
      if (tid < st) red[tid] += red[tid + st];
      __syncthreads();
    }
    if (tid == 0) ls[e] = (ys[e] - red[0]) / L[(size_t)e * CE + e];
    __syncthreads();
  }
  for (int f = tid; f < CE; f += 256) lam[(size_t)b * CE + f] = ls[f];
}

// ---------------- 12: correction scatter ----------------
__global__ void grp_corr(const float* __restrict__ lam, const float* __restrict__ dx4,
                         const int* __restrict__ ei, float* __restrict__ corr) {
  int idx = blockIdx.x * blockDim.x + threadIdx.x;
  if (idx >= CB * CE) return;
  int b = idx / CE, e = idx % CE;
  int i = ei[e], j = ei[CE + e];
  float l2 = 2.0f * lam[idx];
  for (int c = 0; c < 3; ++c) {
    float ce = l2 * dx4[(size_t)idx * 4 + c];
    atomicAdd(&corr[((size_t)b * CN + i) * 3 + c], ce);
    atomicAdd(&corr[((size_t)b * CN + j) * 3 + c], -ce);
  }
}

// ---------------- 13: out = v - corr ----------------
__global__ void grp_out(const float* __restrict__ v, const float* __restrict__ corr,
                        float* __restrict__ out) {
  int idx = blockIdx.x * blockDim.x + threadIdx.x;
  if (idx < CB * CN * 3) out[idx] = v[idx] - corr[idx];
}

extern "C" void kernel_launch(void* const* d_in, const int* in_sizes, int n_in,
                              void* d_out, int out_size, void* d_ws, size_t ws_size,
                              hipStream_t stream) {
  const float* kp   = (const float*)d_in[0];
  const float* ts   = (const float*)d_in[1];
  const float* tok  = (const float*)d_in[2];
  const float* W1   = (const float*)d_in[3];
  const float* b1   = (const float*)d_in[4];
  const float* W2   = (const float*)d_in[5];
  const float* b2   = (const float*)d_in[6];
  const float* tmpl = (const float*)d_in[7];
  const int* groups = (const int*)d_in[8];
  const int* ei     = (const int*)d_in[9];
  float* out = (float*)d_out;
  float* w = (float*)d_ws;

  float* gt     = w + OFF_GTOK;
  float* h      = w + OFF_H;
  float* params = w + OFF_PARAMS;
  float* R      = w + OFF_R;
  float* T      = w + OFF_T;
  float* xsum   = w + OFF_XSUM;
  float* counts = w + OFF_COUNTS;
  float* v      = w + OFF_V;
  float* dx4    = w + OFF_DX4;
  float* rhs    = w + OFF_RHS;
  float* lam    = w + OFF_LAM;
  float* sgn    = w + OFF_SIGN;
  float* corr   = w + OFF_CORR;
  float* A      = w + OFF_A;

  // zero accumulators (xsum+counts contiguous; corr separate)
  grp_zero<<<(49664 + 255) / 256, 256, 0, stream>>>(xsum, 49664);
  grp_zero<<<(CB * CN * 3 + 255) / 256, 256, 0, stream>>>(corr, CB * CN * 3);

  grp_group_mean<<<CB * CG, 256, 0, stream>>>(tok, groups, gt);
  grp_gemm1_wmma<<<64, 256, 0, stream>>>(gt, W1, b1, h);
  grp_gemm2<<<(CB * CG * 9 + 255) / 256, 256, 0, stream>>>(h, W2, b2, params);
  grp_rot<<<(CB * CG + 255) / 256, 256, 0, stream>>>(params, R, T);
  grp_scatter<<<(CB * CG * CS + 255) / 256, 256, 0, stream>>>(R, T, tmpl, groups, xsum, counts);
  grp_vel<<<(CB * CN + 255) / 256, 256, 0, stream>>>(xsum, counts, kp, ts, v);
  grp_edges<<<(CB * CE + 255) / 256, 256, 0, stream>>>(kp, v, ei, dx4, rhs);
  grp_sign<<<(CE * CE + 255) / 256, 256, 0, stream>>>(ei, sgn);
  grp_buildA<<<(CB * 64 * 64) / 8, 256, 0, stream>>>(dx4, sgn, A);
  grp_chol<<<CB, 256, 0, stream>>>(A);
  grp_solve<<<CB, 256, 0, stream>>>(A, rhs, lam);
  grp_corr<<<(CB * CE + 255) / 256, 256, 0, stream>>>(lam, dx4, ei, corr);
  grp_out<<<(CB * CN * 3 + 255) / 256, 256, 0, stream>>>(v, corr, out);

  (void)in_sizes; (void)n_in; (void)out_size; (void)ws_size;
}